// FCOSHead_22058952032749
// MI455X (gfx1250) — compile-verified
//
#include <hip/hip_runtime.h>
#include <cstdint>

#define NLOC 17064
#define NB_TOT ((size_t)16 * NLOC)

typedef __attribute__((ext_vector_type(2))) float v2f;
typedef __attribute__((ext_vector_type(8))) float v8f;

struct InPtrs {
  const float* cls[5];
  const float* cen[5];
  const float* reg[5];
};

__device__ __forceinline__ float sigmoidf_(float x) {
  return 1.0f / (1.0f + expf(-x));
}

// ---------------------------------------------------------------------------
// Kernel A: per-location score/argmax/box decode.
// cls logits streamed via 8-deep async global->LDS pipeline (ASYNCcnt).
// ---------------------------------------------------------------------------
__global__ __launch_bounds__(256) void fcos_score_kernel(
    InPtrs in, float* __restrict__ scoreW, int* __restrict__ clsW,
    float4* __restrict__ boxW) {
  __shared__ float4 sbuf[8][256];
  const int tid = threadIdx.x;
  const int b = blockIdx.x / 20;
  const int tt = blockIdx.x % 20;
  int level, posBase;
  if (tt < 13)      { level = 0; posBase = tt * 1024; }
  else if (tt < 17) { level = 1; posBase = (tt - 13) * 1024; }
  else              { level = tt - 15; posBase = 0; }
  const int HWs[5] = {12800, 3200, 800, 208, 56};
  const int WSH[5] = {7, 6, 5, 4, 3};
  const int STR[5] = {8, 16, 32, 64, 128};
  const int NBA[5] = {0, 12800, 16000, 16800, 17008};
  const int hw = HWs[level], wsh = WSH[level], stride = STR[level], nbase = NBA[level];
  const int pos = posBase + 4 * tid;
  const bool active = pos < hw;
  const int pc = active ? pos : (hw - 4);  // hw is always a multiple of 4

  const float* clsB = in.cls[level] + (size_t)b * 80 * hw;
  const uint32_t ldsBase = (uint32_t)(uintptr_t)(&sbuf[0][tid]);

  auto issue = [&](int c) {
    unsigned long long ga =
        (unsigned long long)(uintptr_t)(clsB + (size_t)c * hw + pc);
    uint32_t la = ldsBase + (uint32_t)((c & 7) * 4096);
    asm volatile("global_load_async_to_lds_b128 %0, %1, off"
                 :: "v"(la), "v"(ga) : "memory");
  };

  float m0 = -3.4e38f, m1 = -3.4e38f, m2 = -3.4e38f, m3 = -3.4e38f;
  int a0 = 0, a1 = 0, a2 = 0, a3 = 0;

  auto consume = [&](int c) {
    float4 v = sbuf[c & 7][tid];
    if (v.x > m0) { m0 = v.x; a0 = c; }
    if (v.y > m1) { m1 = v.y; a1 = c; }
    if (v.z > m2) { m2 = v.z; a2 = c; }
    if (v.w > m3) { m3 = v.w; a3 = c; }
  };

#pragma unroll
  for (int c = 0; c < 8; ++c) issue(c);
  for (int c = 0; c < 72; ++c) {
    issue(c + 8);
    asm volatile("s_wait_asynccnt 7" ::: "memory");  // copies <= c+1 complete
    consume(c);
  }
  asm volatile("s_wait_asynccnt 0" ::: "memory");
  for (int c = 72; c < 80; ++c) consume(c);

  const float* cenB = in.cen[level] + (size_t)b * hw;
  const float* regB = in.reg[level] + (size_t)b * 4 * hw;
  float4 cn = *reinterpret_cast<const float4*>(cenB + pc);
  float4 r0 = *reinterpret_cast<const float4*>(regB + 0 * hw + pc);
  float4 r1 = *reinterpret_cast<const float4*>(regB + 1 * hw + pc);
  float4 r2 = *reinterpret_cast<const float4*>(regB + 2 * hw + pc);
  float4 r3 = *reinterpret_cast<const float4*>(regB + 3 * hw + pc);

  if (active) {
    const int half = stride >> 1;
    const int wmask = (1 << wsh) - 1;
#define EMIT(J, MV, AV, CC, R0, R1, R2, R3)                                   \
    {                                                                         \
      int idx = pc + (J);                                                     \
      float cx = (float)((idx & wmask) * stride + half);                      \
      float cy = (float)((idx >> wsh) * stride + half);                       \
      float sc = sqrtf(sigmoidf_(MV) * sigmoidf_(CC));                        \
      size_t row = (size_t)b * NLOC + nbase + idx;                            \
      scoreW[row] = sc;                                                       \
      clsW[row] = (AV) + 1;                                                   \
      boxW[row] = make_float4(cx - (R0), cy - (R1), cx + (R2), cy + (R3));    \
    }
    EMIT(0, m0, a0, cn.x, r0.x, r1.x, r2.x, r3.x)
    EMIT(1, m1, a1, cn.y, r0.y, r1.y, r2.y, r3.y)
    EMIT(2, m2, a2, cn.z, r0.z, r1.z, r2.z, r3.z)
    EMIT(3, m3, a3, cn.w, r0.w, r1.w, r2.w, r3.w)
#undef EMIT
  }
}

// ---------------------------------------------------------------------------
// Kernel B: exact top-1000 per batch, sorted descending (ties -> lower index).
// Bitwise binary search on unique 64-bit keys + LDS bitonic sort of 1024.
// ---------------------------------------------------------------------------
__global__ __launch_bounds__(1024) void fcos_topk_kernel(
    const float* __restrict__ scoreW, const int* __restrict__ clsW,
    const float4* __restrict__ boxW, float* __restrict__ topS,
    int* __restrict__ topC, float4* __restrict__ topB,
    float* __restrict__ maxcW) {
  const int b = blockIdx.x, tid = threadIdx.x;
  __shared__ unsigned cnt;
  __shared__ unsigned long long skey[1024];
  __shared__ float red[1024];

  unsigned long long keys[17];
#pragma unroll
  for (int r = 0; r < 17; ++r) {
    int n = tid + r * 1024;
    unsigned long long k = 0ull;
    if (n < NLOC) {
      unsigned sb = __float_as_uint(scoreW[(size_t)b * NLOC + n]);
      k = ((unsigned long long)sb << 32) | (unsigned)(0xFFFFFFFFu - (unsigned)n);
    }
    keys[r] = k;
  }

  unsigned long long thr = 0ull;
  for (int bit = 63; bit >= 0; --bit) {
    unsigned long long cand = thr | (1ull << bit);
    if (tid == 0) cnt = 0u;
    __syncthreads();
    int local = 0;
#pragma unroll
    for (int r = 0; r < 17; ++r) local += (keys[r] >= cand) ? 1 : 0;
    for (int off = 16; off > 0; off >>= 1) local += __shfl_down(local, off, 32);
    if ((tid & 31) == 0) atomicAdd(&cnt, (unsigned)local);
    __syncthreads();
    if (cnt >= 1000u) thr = cand;
    __syncthreads();
  }
  // thr is now exactly the 1000th-largest key; count(>=thr) == 1000.

  skey[tid] = 0ull;
  if (tid == 0) cnt = 0u;
  __syncthreads();
#pragma unroll
  for (int r = 0; r < 17; ++r) {
    if (keys[r] >= thr) {
      unsigned slot = atomicAdd(&cnt, 1u);
      skey[slot] = keys[r];
    }
  }
  __syncthreads();

  // bitonic sort, descending
  for (int k = 2; k <= 1024; k <<= 1) {
    for (int j = k >> 1; j > 0; j >>= 1) {
      int ixj = tid ^ j;
      if (ixj > tid) {
        bool up = ((tid & k) == 0);
        unsigned long long a = skey[tid], c = skey[ixj];
        bool sw = up ? (a < c) : (a > c);
        if (sw) { skey[tid] = c; skey[ixj] = a; }
      }
      __syncthreads();
    }
  }

  unsigned long long k = skey[tid];
  int n = (int)(0xFFFFFFFFu - (unsigned)(k & 0xFFFFFFFFull));
  float s = __uint_as_float((unsigned)(k >> 32));
  float4 box = make_float4(0.f, 0.f, 0.f, 0.f);
  int c = 0;
  if (tid < 1000) {
    c = clsW[(size_t)b * NLOC + n];
    box = boxW[(size_t)b * NLOC + n];
  }
  topS[b * 1024 + tid] = (tid < 1000) ? s : 0.f;
  topC[b * 1024 + tid] = c;
  topB[b * 1024 + tid] = box;

  float m = fmaxf(fmaxf(fabsf(box.x), fabsf(box.y)),
                  fmaxf(fabsf(box.z), fabsf(box.w)));
  red[tid] = (tid < 1000) ? m : 0.f;
  __syncthreads();
  for (int s2 = 512; s2 > 0; s2 >>= 1) {
    if (tid < s2) red[tid] = fmaxf(red[tid], red[tid + s2]);
    __syncthreads();
  }
  if (tid == 0) maxcW[b] = red[0] + 1.0f;
}

// ---------------------------------------------------------------------------
// Kernel C1: build suppression bitmask. mask[b][i][w] bit j%32: iou(i,j)>thr
// and j>i. Pairwise area sums via v_wmma_f32_16x16x4_f32 ([a_i,1]x[1;a_j]).
// IoU test is division-free: inter/(union+eps) > 0.5  <=>  2*inter > union+eps
// (union + eps is strictly positive).
// ---------------------------------------------------------------------------
__global__ __launch_bounds__(1024) void fcos_mask_kernel(
    const float* __restrict__ topS, const int* __restrict__ topC,
    const float4* __restrict__ topB, const float* __restrict__ maxcW,
    unsigned* __restrict__ maskW) {
  const int b = blockIdx.x, tid = threadIdx.x;
  __shared__ float4 bbS[1024];
  __shared__ float areaS[1024];
  const float maxc = maxcW[b];
  {
    float4 bx = topB[b * 1024 + tid];
    float off = (float)topC[b * 1024 + tid] * maxc;
    bbS[tid] = make_float4(bx.x + off, bx.y + off, bx.z + off, bx.w + off);
    areaS[tid] = (bx.z - bx.x) * (bx.w - bx.y);  // offset-invariant
  }
  __syncthreads();

  const int lane = tid & 31;
  const int gw = blockIdx.y * 32 + (tid >> 5);  // grid.y == 4 -> 128 waves/batch
  unsigned* maskB = maskW + (size_t)b * 1024 * 32;

  for (int tile = gw; tile < 63 * 32; tile += 128) {
    const int rt = tile >> 5, ct = tile & 31;
    const int i0 = rt * 16, j0 = ct * 32;
    if (j0 + 31 <= i0) {  // entirely j<=i: all-zero words, still must write
      if (lane < 16) maskB[(size_t)(i0 + lane) * 32 + ct] = 0u;
      continue;
    }
    // A(16x4): K0=area_i, K1=1 (lanes 0..15 hold K0,K1; lanes 16..31 K2,K3=0)
    v2f av;  av.x  = (lane < 16) ? areaS[i0 + lane] : 0.f;
             av.y  = (lane < 16) ? 1.f : 0.f;
    v2f bv0; bv0.x = (lane < 16) ? 1.f : 0.f;
             bv0.y = (lane < 16) ? areaS[j0 + lane] : 0.f;
    v2f bv1; bv1.x = bv0.x;
             bv1.y = (lane < 16) ? areaS[j0 + 16 + lane] : 0.f;
    v8f c0 = {};
#if __has_builtin(__builtin_amdgcn_wmma_f32_16x16x4_f32)
    v8f s0 = __builtin_amdgcn_wmma_f32_16x16x4_f32(false, av, false, bv0,
                                                   (short)0, c0, false, false);
    v8f s1 = __builtin_amdgcn_wmma_f32_16x16x4_f32(false, av, false, bv1,
                                                   (short)0, c0, false, false);
#define AREASUM0(i, j, comp) (s0[comp])
#define AREASUM1(i, j, comp) (s1[comp])
#else
#define AREASUM0(i, j, comp) (areaS[i] + areaS[j])
#define AREASUM1(i, j, comp) (areaS[i] + areaS[j])
#endif
    const int Mb = (lane >> 4) * 8;
    const int N = lane & 15;
    unsigned bal0[8], bal1[8];
#pragma unroll
    for (int cc = 0; cc < 8; ++cc) {
      const int i = i0 + Mb + cc;
      float4 bi = bbS[i];
      {
        const int j = j0 + N;
        float4 bj = bbS[j];
        float iw = fmaxf(fminf(bi.z, bj.z) - fmaxf(bi.x, bj.x), 0.f);
        float ih = fmaxf(fminf(bi.w, bj.w) - fmaxf(bi.y, bj.y), 0.f);
        float inter = iw * ih;
        // iou > 0.5 with positive denominator, division-free:
        bool pred = (2.0f * inter > (AREASUM0(i, j, cc) - inter + 1e-9f)) &&
                    (j > i) && (j < 1000) && (i < 1000);
#if __has_builtin(__builtin_amdgcn_ballot_w32)
        bal0[cc] = __builtin_amdgcn_ballot_w32(pred);
#else
        bal0[cc] = (unsigned)__ballot(pred);
#endif
      }
      {
        const int j = j0 + 16 + N;
        float4 bj = bbS[j];
        float iw = fmaxf(fminf(bi.z, bj.z) - fmaxf(bi.x, bj.x), 0.f);
        float ih = fmaxf(fminf(bi.w, bj.w) - fmaxf(bi.y, bj.y), 0.f);
        float inter = iw * ih;
        bool pred = (2.0f * inter > (AREASUM1(i, j, cc) - inter + 1e-9f)) &&
                    (j > i) && (j < 1000) && (i < 1000);
#if __has_builtin(__builtin_amdgcn_ballot_w32)
        bal1[cc] = __builtin_amdgcn_ballot_w32(pred);
#else
        bal1[cc] = (unsigned)__ballot(pred);
#endif
      }
    }
#undef AREASUM0
#undef AREASUM1
    if (lane == 0) {
#pragma unroll
      for (int cc = 0; cc < 8; ++cc) {
        // ballot low 16 = rows i0+cc (lanes 0..15), high 16 = rows i0+8+cc
        maskB[(size_t)(i0 + cc) * 32 + ct] =
            (bal0[cc] & 0xFFFFu) | ((bal1[cc] & 0xFFFFu) << 16);
        maskB[(size_t)(i0 + 8 + cc) * 32 + ct] =
            (bal0[cc] >> 16) | ((bal1[cc] >> 16) << 16);
      }
    }
  }
}

// ---------------------------------------------------------------------------
// Kernel C2: sequential suppression (single wave, keep-bits in registers),
// then write all outputs. Prefetch hides mask-row latency.
// ---------------------------------------------------------------------------
__global__ __launch_bounds__(256) void fcos_nms_out_kernel(
    const float* __restrict__ topS, const int* __restrict__ topC,
    const float4* __restrict__ topB, const unsigned* __restrict__ maskW,
    float* __restrict__ out) {
  const int b = blockIdx.x, tid = threadIdx.x;
  __shared__ unsigned keepW[32];
  const unsigned* maskB = maskW + (size_t)b * 1024 * 32;

  if (tid < 32) {
    unsigned kw = 0u;
    for (int bit = 0; bit < 32; ++bit) {
      int j = tid * 32 + bit;
      if (j < 1000 && topS[b * 1024 + j] >= 0.5f) kw |= (1u << bit);
    }
    for (int i = 0; i < 1000; ++i) {
      __builtin_prefetch(&maskB[(size_t)(i + 8) * 32 + tid], 0, 0);
      unsigned kiWord = __shfl(kw, i >> 5, 32);
      if (kiWord & (1u << (i & 31))) {
        kw &= ~maskB[(size_t)i * 32 + tid];
      }
    }
    keepW[tid] = kw;
  }
  __syncthreads();

  for (int t = tid; t < 1000; t += 256) {
    bool keep = (keepW[t >> 5] >> (t & 31)) & 1u;
    float s = topS[b * 1024 + t];
    int c = topC[b * 1024 + t];
    float4 bx = topB[b * 1024 + t];
    out[b * 1000 + t] = keep ? s : 0.f;
    out[16 * 1000 + b * 1000 + t] = keep ? (float)c : 0.f;
    float4 z = make_float4(0.f, 0.f, 0.f, 0.f);
    *reinterpret_cast<float4*>(out + 32 * 1000 + ((size_t)b * 1000 + t) * 4) =
        keep ? bx : z;
  }
}

// ---------------------------------------------------------------------------
extern "C" void kernel_launch(void* const* d_in, const int* in_sizes, int n_in,
                              void* d_out, int out_size, void* d_ws,
                              size_t ws_size, hipStream_t stream) {
  (void)in_sizes; (void)n_in; (void)out_size; (void)ws_size;
  InPtrs ip;
  for (int i = 0; i < 5; ++i) {
    ip.cls[i] = (const float*)d_in[3 * i + 0];
    ip.cen[i] = (const float*)d_in[3 * i + 1];
    ip.reg[i] = (const float*)d_in[3 * i + 2];
  }
  float* wsf = (float*)d_ws;
  float*    scoreW = wsf;                                        // B*NLOC
  int*      clsW   = (int*)(wsf + NB_TOT);                       // B*NLOC
  float4*   boxW   = (float4*)(wsf + 2 * NB_TOT);                // B*NLOC*4
  float*    topS   = wsf + 6 * NB_TOT;                           // B*1024
  int*      topC   = (int*)(wsf + 6 * NB_TOT + 16 * 1024);       // B*1024
  float4*   topB   = (float4*)(wsf + 6 * NB_TOT + 32 * 1024);    // B*1024*4
  float*    maxcW  = wsf + 6 * NB_TOT + 96 * 1024;               // B
  unsigned* maskW  = (unsigned*)(wsf + 6 * NB_TOT + 96 * 1024 + 16); // B*1024*32
  float* out = (float*)d_out;

  fcos_score_kernel<<<dim3(320), dim3(256), 0, stream>>>(ip, scoreW, clsW, boxW);
  fcos_topk_kernel<<<dim3(16), dim3(1024), 0, stream>>>(scoreW, clsW, boxW,
                                                        topS, topC, topB, maxcW);
  fcos_mask_kernel<<<dim3(16, 4), dim3(1024), 0, stream>>>(topS, topC, topB,
                                                           maxcW, maskW);
  fcos_nms_out_kernel<<<dim3(16), dim3(256), 0, stream>>>(topS, topC, topB,
                                                          maskW, out);
}